// ODESolver_15006615732354
// MI455X (gfx1250) — compile-verified
//
#include <hip/hip_runtime.h>
#include <hip/hip_bf16.h>
#include <math.h>

typedef __attribute__((ext_vector_type(2))) float v2f;
typedef __attribute__((ext_vector_type(8))) float v8f;

// gfx1250 has a hardware V_TANH_F32 (TRANS pipe). Try both plausible builtin
// spellings; otherwise force the instruction via inline asm. The trailing
// v_nop provides the gap the ISA requires between a TRANS op and
// a use of its result (CDNA5 ISA 7.4 #3).
#if __has_builtin(__builtin_amdgcn_tanh_f32)
#define TANHF(v) __builtin_amdgcn_tanh_f32(v)
#elif __has_builtin(__builtin_amdgcn_tanhf)
#define TANHF(v) __builtin_amdgcn_tanhf(v)
#else
static __device__ __forceinline__ float hw_tanhf(float x) {
  float r;
  asm volatile("v_tanh_f32 %0, %1\n\tv_nop" : "=v"(r) : "v"(x));
  return r;
}
#define TANHF(v) hw_tanhf(v)
#endif

#define D_ 32
#define H_ 50
#define K_ 5
#define HP 64                 // H padded to WMMA tile multiple
#define ROWS_PER_WAVE 32      // 2 M-tiles per wave: B-fragments reused twice
#define ROWS_PER_BLOCK 256    // 8 waves x 32 rows
#define MAXT 64

// ---------------------------------------------------------------------------
// Workspace layout (bytes):
//   [0..3]     u32 presence mask of row-min values {0..9}
//   [4..7]     i32 T (number of grid times)
//   [8..47]    i32 map10[10]  (value -> grid index)
//   [48..303]  f32 dts[MAXT]
//   [512.. )   u8  rm[B]      (per-row min value)
// ---------------------------------------------------------------------------

__global__ __launch_bounds__(256) void rowmin_mask_kernel(
    const int* __restrict__ t_int, unsigned char* __restrict__ rm,
    unsigned int* __restrict__ mask, int B) {
  __shared__ unsigned int sm;
  if (threadIdx.x == 0) sm = 0u;
  __syncthreads();
  int b = blockIdx.x * 256 + threadIdx.x;
  if (b < B) {
    int m = 0x7fffffff;
    for (int k = 0; k < K_; ++k) {
      int v = t_int[(size_t)b * K_ + k];
      if (v < 0) v = -v;
      if (v < m) m = v;
    }
    rm[b] = (unsigned char)m;
    atomicOr(&sm, 1u << (m & 31));
  }
  __syncthreads();
  if (threadIdx.x == 0) atomicOr(mask, sm);
}

// Single-thread grid construction mirroring _split_time (numpy float64 math).
__global__ void build_grid_kernel(int* __restrict__ prm, float* __restrict__ dts) {
  if (threadIdx.x != 0 || blockIdx.x != 0) return;
  unsigned int mk = ((const unsigned int*)prm)[0];
  double res[10];
  int nres = 0;
  for (int v = 0; v < 10; ++v)
    if (mk & (1u << v)) res[nres++] = (double)v / 10.0;
  if (nres == 0) { res[nres++] = 0.0; }  // degenerate guard
  double max_time = res[nres - 1];
  if (max_time < 1e-4) max_time = 1e-4;
  double step = (0.1 <= max_time) ? 0.1 : max_time / 20.0;
  int n = (int)(max_time / step);  // python int() truncation
  double buf[MAXT];
  int nb = 0;
  for (int i = 0; i < nres; ++i) buf[nb++] = res[i];
  if (n == 1) {
    buf[nb++] = 0.0;
  } else if (n > 1) {
    double delta = max_time / (double)(n - 1);  // np.linspace endpoint=True
    for (int i = 0; i < n - 1; ++i) buf[nb++] = delta * (double)i;
    buf[nb++] = max_time;  // linspace sets last element exactly
  }
  // sort (insertion) + exact-equality dedupe == np.unique
  for (int i = 1; i < nb; ++i) {
    double key = buf[i];
    int j = i - 1;
    while (j >= 0 && buf[j] > key) { buf[j + 1] = buf[j]; --j; }
    buf[j + 1] = key;
  }
  double times[MAXT];
  int T = 0;
  for (int i = 0; i < nb; ++i)
    if (T == 0 || buf[i] != times[T - 1]) times[T++] = buf[i];
  prm[1] = T;
  for (int v = 0; v < 10; ++v) {
    double tv = (double)v / 10.0;
    int idx = 0;
    for (int i = 0; i < T; ++i)
      if (times[i] == tv) { idx = i; break; }
    prm[2 + v] = idx;
  }
  for (int i = 0; i < MAXT; ++i) dts[i] = 0.0f;
  for (int i = 0; i + 1 < T; ++i)
    dts[i] = (float)times[i + 1] - (float)times[i];  // f32 diff like np.diff(f32)
}

// ---------------------------------------------------------------------------
// Main kernel: each wave integrates a 32-row tile (2 WMMA M-tiles), so every
// weight B-fragment load feeds two v_wmma_f32_16x16x4_f32 ops.
// Fragment layouts (wave32, per CDNA5 ISA 7.12.2):
//   A 16x4:  lane<16 -> row=lane, {K0,K1}; lane>=16 -> row=lane-16, {K2,K3}
//   B 4x16:  lane<16 -> col=lane, {K0,K1}; lane>=16 -> col=lane-16, {K2,K3}
//   C/D:     lane<16 -> col=lane, vgpr v = row v; lane>=16 -> row 8+v
// ---------------------------------------------------------------------------
__global__ __launch_bounds__(256) void ode_euler_wmma_kernel(
    const float* __restrict__ x, const unsigned char* __restrict__ rm,
    const int* __restrict__ prm, const float* __restrict__ dts,
    const float* __restrict__ W1, const float* __restrict__ b1,
    const float* __restrict__ W2, const float* __restrict__ b2,
    const float* __restrict__ W3, const float* __restrict__ b3,
    float* __restrict__ out) {
  __shared__ float sW1[HP * D_];        // col-major [n][k], k-stride 32
  __shared__ float sW2[HP * HP];        // col-major [n][k], k-stride 64
  __shared__ float sW3[D_ * HP];        // col-major [n][k], k-stride 64
  __shared__ float sb1[HP], sb2[HP], sb3[D_];
  __shared__ float yb[8 * ROWS_PER_WAVE * D_];   // per-wave y tile, 32x32
  __shared__ float hbuf[8 * ROWS_PER_WAVE * HP]; // per-wave h tile, 32x64
  __shared__ int ridx[ROWS_PER_BLOCK];

  const int tid = threadIdx.x;
  const int wave = tid >> 5;
  const int lane = tid & 31;
  const int half = lane >> 4;
  const int mr = lane & 15;
  const int rowbase = blockIdx.x * ROWS_PER_BLOCK;

  // stage weights (transposed to col-major, zero-padded) + biases into LDS
  for (int i = tid; i < HP * D_; i += 256) {
    int n = i >> 5, k = i & (D_ - 1);
    sW1[i] = (n < H_) ? W1[k * H_ + n] : 0.0f;
  }
  for (int i = tid; i < HP * HP; i += 256) {
    int n = i >> 6, k = i & (HP - 1);
    sW2[i] = (n < H_ && k < H_) ? W2[k * H_ + n] : 0.0f;
  }
  for (int i = tid; i < D_ * HP; i += 256) {
    int n = i >> 6, k = i & (HP - 1);
    sW3[i] = (k < H_) ? W3[k * D_ + n] : 0.0f;
  }
  if (tid < HP) {
    sb1[tid] = (tid < H_) ? b1[tid] : 0.0f;
    sb2[tid] = (tid < H_) ? b2[tid] : 0.0f;
  }
  if (tid < D_) sb3[tid] = b3[tid];
  ridx[tid] = prm[2 + rm[rowbase + tid]];

  {  // coalesced float4 load of the x tile into per-wave y buffers
    const float4* x4 = reinterpret_cast<const float4*>(x + (size_t)rowbase * D_);
    float4* yb4 = reinterpret_cast<float4*>(yb);
    for (int i = tid; i < ROWS_PER_BLOCK * (D_ / 4); i += 256) {
      int row = i >> 3, c4 = i & 7;
      int w = row >> 5, m = row & 31;
      yb4[w * (ROWS_PER_WAVE * 8) + m * 8 + c4] = x4[i];
    }
  }
  __syncthreads();

  const int T = prm[1];
  float* Y = yb + wave * (ROWS_PER_WAVE * D_);
  float* Hs = hbuf + wave * (ROWS_PER_WAVE * HP);

  // loop-invariant bias fragments (broadcast per C/D column)
  float bv1[4], bv2[4], bv3[2];
#pragma unroll
  for (int nt = 0; nt < 4; ++nt) {
    bv1[nt] = sb1[nt * 16 + mr];
    bv2[nt] = sb2[nt * 16 + mr];
  }
#pragma unroll
  for (int nt = 0; nt < 2; ++nt) bv3[nt] = sb3[nt * 16 + mr];

  // y kept in registers in C/D-fragment layout: [mtile][ntile]
  v8f yreg[2][2];
#pragma unroll
  for (int mt = 0; mt < 2; ++mt)
#pragma unroll
    for (int nt = 0; nt < 2; ++nt)
#pragma unroll
      for (int v = 0; v < 8; ++v)
        yreg[mt][nt][v] = Y[(mt * 16 + v + 8 * half) * D_ + nt * 16 + mr];

  // snapshot: each lane owns one full row of its wave's 32-row tile
  const int myidx = ridx[wave * ROWS_PER_WAVE + lane];
  const float4* ysnap = reinterpret_cast<const float4*>(Y + lane * D_);
  float4* osnap = reinterpret_cast<float4*>(
      out + (size_t)(rowbase + wave * ROWS_PER_WAVE + lane) * D_);

  if (myidx == 0) {
#pragma unroll
    for (int q = 0; q < 8; ++q) osnap[q] = ysnap[q];
  }

  for (int s = 0; s < T - 1; ++s) {
    const float dt = dts[s];

    // ---- layer 1: h1 = tanh(y @ W1 + b1)   (32x32 @ 32x64) ----
    v8f acc[2][4];
#pragma unroll
    for (int mt = 0; mt < 2; ++mt)
#pragma unroll
      for (int nt = 0; nt < 4; ++nt)
#pragma unroll
        for (int v = 0; v < 8; ++v) acc[mt][nt][v] = bv1[nt];
#pragma unroll
    for (int kt = 0; kt < D_ / 4; ++kt) {
      v2f a[2];
#pragma unroll
      for (int mt = 0; mt < 2; ++mt)
        a[mt] = *reinterpret_cast<const v2f*>(Y + (mt * 16 + mr) * D_ +
                                              kt * 4 + 2 * half);
#pragma unroll
      for (int nt = 0; nt < 4; ++nt) {
        v2f b = *reinterpret_cast<const v2f*>(sW1 + (nt * 16 + mr) * D_ +
                                              kt * 4 + 2 * half);
#pragma unroll
        for (int mt = 0; mt < 2; ++mt)
          acc[mt][nt] = __builtin_amdgcn_wmma_f32_16x16x4_f32(
              false, a[mt], false, b, (short)0, acc[mt][nt], false, false);
      }
    }
#pragma unroll
    for (int mt = 0; mt < 2; ++mt)
#pragma unroll
      for (int nt = 0; nt < 4; ++nt)
#pragma unroll
        for (int v = 0; v < 8; ++v)
          Hs[(mt * 16 + v + 8 * half) * HP + nt * 16 + mr] =
              TANHF(acc[mt][nt][v]);

    // ---- layer 2: h2 = tanh(h1 @ W2 + b2)   (32x64 @ 64x64) ----
#pragma unroll
    for (int mt = 0; mt < 2; ++mt)
#pragma unroll
      for (int nt = 0; nt < 4; ++nt)
#pragma unroll
        for (int v = 0; v < 8; ++v) acc[mt][nt][v] = bv2[nt];
#pragma unroll
    for (int kt = 0; kt < HP / 4; ++kt) {
      v2f a[2];
#pragma unroll
      for (int mt = 0; mt < 2; ++mt)
        a[mt] = *reinterpret_cast<const v2f*>(Hs + (mt * 16 + mr) * HP +
                                              kt * 4 + 2 * half);
#pragma unroll
      for (int nt = 0; nt < 4; ++nt) {
        v2f b = *reinterpret_cast<const v2f*>(sW2 + (nt * 16 + mr) * HP +
                                              kt * 4 + 2 * half);
#pragma unroll
        for (int mt = 0; mt < 2; ++mt)
          acc[mt][nt] = __builtin_amdgcn_wmma_f32_16x16x4_f32(
              false, a[mt], false, b, (short)0, acc[mt][nt], false, false);
      }
    }
    // all Hs reads above complete before these overwrites (same-wave DS order)
#pragma unroll
    for (int mt = 0; mt < 2; ++mt)
#pragma unroll
      for (int nt = 0; nt < 4; ++nt)
#pragma unroll
        for (int v = 0; v < 8; ++v)
          Hs[(mt * 16 + v + 8 * half) * HP + nt * 16 + mr] =
              TANHF(acc[mt][nt][v]);

    // ---- layer 3 + Euler: y += dt * tanh(h2 @ W3 + b3)  (32x64 @ 64x32) ----
    v8f acc3[2][2];
#pragma unroll
    for (int mt = 0; mt < 2; ++mt)
#pragma unroll
      for (int nt = 0; nt < 2; ++nt)
#pragma unroll
        for (int v = 0; v < 8; ++v) acc3[mt][nt][v] = bv3[nt];
#pragma unroll
    for (int kt = 0; kt < HP / 4; ++kt) {
      v2f a[2];
#pragma unroll
      for (int mt = 0; mt < 2; ++mt)
        a[mt] = *reinterpret_cast<const v2f*>(Hs + (mt * 16 + mr) * HP +
                                              kt * 4 + 2 * half);
#pragma unroll
      for (int nt = 0; nt < 2; ++nt) {
        v2f b = *reinterpret_cast<const v2f*>(sW3 + (nt * 16 + mr) * HP +
                                              kt * 4 + 2 * half);
#pragma unroll
        for (int mt = 0; mt < 2; ++mt)
          acc3[mt][nt] = __builtin_amdgcn_wmma_f32_16x16x4_f32(
              false, a[mt], false, b, (short)0, acc3[mt][nt], false, false);
      }
    }
#pragma unroll
    for (int mt = 0; mt < 2; ++mt)
#pragma unroll
      for (int nt = 0; nt < 2; ++nt)
#pragma unroll
        for (int v = 0; v < 8; ++v) {
          float f = TANHF(acc3[mt][nt][v]);
          float yn = fmaf(dt, f, yreg[mt][nt][v]);
          yreg[mt][nt][v] = yn;
          Y[(mt * 16 + v + 8 * half) * D_ + nt * 16 + mr] = yn;
        }

    if (myidx == s + 1) {
#pragma unroll
      for (int q = 0; q < 8; ++q) osnap[q] = ysnap[q];
    }
  }
}

extern "C" void kernel_launch(void* const* d_in, const int* in_sizes, int n_in,
                              void* d_out, int out_size, void* d_ws,
                              size_t ws_size, hipStream_t stream) {
  const float* x = (const float*)d_in[0];
  const int* t_int = (const int*)d_in[1];
  const float* W1 = (const float*)d_in[2];
  const float* b1 = (const float*)d_in[3];
  const float* W2 = (const float*)d_in[4];
  const float* b2 = (const float*)d_in[5];
  const float* W3 = (const float*)d_in[6];
  const float* b3 = (const float*)d_in[7];
  float* out = (float*)d_out;

  const int B = in_sizes[0] / D_;

  unsigned char* ws8 = (unsigned char*)d_ws;
  unsigned int* mask = (unsigned int*)ws8;
  int* prm = (int*)ws8;                 // [0]=mask, [1]=T, [2..11]=map10
  float* dts = (float*)(ws8 + 48);
  unsigned char* rm = ws8 + 512;

  hipMemsetAsync(mask, 0, sizeof(unsigned int), stream);
  rowmin_mask_kernel<<<(B + 255) / 256, 256, 0, stream>>>(t_int, rm, mask, B);
  build_grid_kernel<<<1, 1, 0, stream>>>(prm, dts);
  ode_euler_wmma_kernel<<<B / ROWS_PER_BLOCK, 256, 0, stream>>>(
      x, rm, prm, dts, W1, b1, W2, b2, W3, b3, out);
}